// RGCNSampling_56212531970344
// MI455X (gfx1250) — compile-verified
//
#include <hip/hip_runtime.h>
#include <hip/hip_bf16.h>

// ---------------------------------------------------------------------------
// RGCN-style 3-layer multi-relation GAT for MI455X (gfx1250, wave32).
//  * hd = x@W_dst is algebraically eliminated: er = x . (W_dst @ attn_r).
//  * hs = x@W_src computed with V_WMMA_F32_16X16X4_F32 (fp32 matrix pipe).
//  * GEMM staging uses GLOBAL_LOAD_ASYNC_TO_LDS_B128 (ASYNCcnt) with
//    double-buffered W K-panels: DMA of panel kp+1 overlaps WMMA on panel kp.
//  * edge softmax: atomic max (order-preserving uint encoding) + atomic fadd.
//  * message scatter: wave32 per edge, float4/lane, global_atomic_add_f32.
//  * relu+LayerNorm fused, wave per node, __shfl_xor wave32 reductions.
// ---------------------------------------------------------------------------

typedef __attribute__((ext_vector_type(2))) float v2f;
typedef __attribute__((ext_vector_type(8))) float v8f;

constexpr int NN = 50000;   // nodes
constexpr int DD = 128;     // feature dim
constexpr int RR = 4;       // relations
constexpr int EE = 200000;  // edges per relation
constexpr int LL = 3;       // layers
constexpr float LN_EPS    = 1e-5f;
constexpr float NEG_SLOPE = 0.2f;

// ---------------- async global -> LDS helpers (CDNA5, ASYNCcnt) ------------
// LDS byte offset = low 32 bits of a generic pointer into __shared__ space
// (flat LDS aperture keeps the workgroup-relative offset in addr[31:0]).
__device__ __forceinline__ void async_copy_b128(void* lds_dst, const void* gsrc) {
  unsigned loff = (unsigned)(uintptr_t)lds_dst;
  asm volatile("global_load_async_to_lds_b128 %0, %1, off"
               :: "v"(loff), "v"(gsrc)
               : "memory");
}
__device__ __forceinline__ void wait_async0() {
#if __has_builtin(__builtin_amdgcn_s_wait_asynccnt)
  __builtin_amdgcn_s_wait_asynccnt(0);
#else
  asm volatile("s_wait_asynccnt 0" ::: "memory");
#endif
}

// ---------------- order-preserving float <-> uint for atomic max -----------
__device__ __forceinline__ unsigned encf(float f) {
  int i = __float_as_int(f);
  return (unsigned)(i ^ ((i >> 31) | 0x80000000));
}
__device__ __forceinline__ float decf(unsigned u) {
  int i = (u & 0x80000000u) ? (int)(u ^ 0x80000000u) : (int)(~u);
  return __int_as_float(i);
}

// ---------------- tiny: w_r = W_src@attn_l, u_r = W_dst@attn_r -------------
__global__ __launch_bounds__(DD)
void attn_vec_kernel(const float* __restrict__ Wsrc, const float* __restrict__ Wdst,
                     const float* __restrict__ al, const float* __restrict__ ar,
                     float* __restrict__ wvec, float* __restrict__ uvec) {
  const int r = blockIdx.x;     // relation
  const int k = threadIdx.x;    // row of W
  const float* Ws = Wsrc + (size_t)r * DD * DD + (size_t)k * DD;
  const float* Wd = Wdst + (size_t)r * DD * DD + (size_t)k * DD;
  const float* a0 = al + r * DD;
  const float* a1 = ar + r * DD;
  float s0 = 0.f, s1 = 0.f;
  for (int j = 0; j < DD; ++j) { s0 += Ws[j] * a0[j]; s1 += Wd[j] * a1[j]; }
  wvec[r * DD + k] = s0;
  uvec[r * DD + k] = s1;
}

// ---------------- el[r,i] = x_i . wvec_r ; er[r,i] = x_i . uvec_r ----------
__global__ __launch_bounds__(256)
void eler_kernel(const float* __restrict__ x, const float* __restrict__ wvec,
                 const float* __restrict__ uvec, float* __restrict__ el,
                 float* __restrict__ er) {
  const int wave = threadIdx.x >> 5;
  const int lane = threadIdx.x & 31;
  const int node = blockIdx.x * 8 + wave;
  if (node >= NN) return;
  const float4 xv = reinterpret_cast<const float4*>(x + (size_t)node * DD)[lane];
  float accl[RR], accr[RR];
#pragma unroll
  for (int r = 0; r < RR; ++r) {
    const float4 wv = reinterpret_cast<const float4*>(wvec + r * DD)[lane];
    const float4 uv = reinterpret_cast<const float4*>(uvec + r * DD)[lane];
    accl[r] = xv.x * wv.x + xv.y * wv.y + xv.z * wv.z + xv.w * wv.w;
    accr[r] = xv.x * uv.x + xv.y * uv.y + xv.z * uv.z + xv.w * uv.w;
  }
#pragma unroll
  for (int r = 0; r < RR; ++r) {
    float sl = accl[r], sr = accr[r];
#pragma unroll
    for (int off = 16; off; off >>= 1) {
      sl += __shfl_xor(sl, off, 32);
      sr += __shfl_xor(sr, off, 32);
    }
    if (lane == 0) { el[r * NN + node] = sl; er[r * NN + node] = sr; }
  }
}

// ---------------- init acc with ntype_bias + sum_r gat_bias ----------------
__global__ __launch_bounds__(256)
void init_acc_kernel(const float* __restrict__ ntb, const float* __restrict__ gb,
                     float* __restrict__ acc) {
  const int idx = blockIdx.x * 256 + threadIdx.x;
  if (idx >= NN * DD) return;
  const int j = idx & (DD - 1);
  acc[idx] = ntb[j] + gb[0 * DD + j] + gb[1 * DD + j] + gb[2 * DD + j] + gb[3 * DD + j];
}

__global__ __launch_bounds__(256)
void init_stats_kernel(unsigned* __restrict__ mEnc, float* __restrict__ denom) {
  const int idx = blockIdx.x * 256 + threadIdx.x;
  if (idx >= RR * NN) return;
  mEnc[idx] = 0u;        // encodes "smallest possible" -> acts as -inf
  denom[idx] = 0.f;
}

// ---------------- edge pass 1: e = lrelu(el[src]+er[dst]); seg max ---------
__global__ __launch_bounds__(256)
void edge_max_kernel(const int* __restrict__ esrc, const int* __restrict__ edst,
                     const float* __restrict__ el, const float* __restrict__ er,
                     float* __restrict__ eEdge, unsigned* __restrict__ mEnc) {
  const int idx = blockIdx.x * 256 + threadIdx.x;
  if (idx >= RR * EE) return;
  const int r = idx / EE;
  const int s = esrc[idx];
  const int d = edst[idx];
  float ev = el[r * NN + s] + er[r * NN + d];
  ev = (ev >= 0.f) ? ev : NEG_SLOPE * ev;
  eEdge[idx] = ev;
  atomicMax(&mEnc[r * NN + d], encf(ev));
}

// ---------------- edge pass 2: a = exp(e - m[dst]); seg sum ----------------
__global__ __launch_bounds__(256)
void edge_exp_kernel(const int* __restrict__ edst, const unsigned* __restrict__ mEnc,
                     float* __restrict__ eEdge, float* __restrict__ denom) {
  const int idx = blockIdx.x * 256 + threadIdx.x;
  if (idx >= RR * EE) return;
  const int r = idx / EE;
  const int d = edst[idx];
  const float a = __expf(eEdge[idx] - decf(mEnc[r * NN + d]));
  eEdge[idx] = a;
  unsafeAtomicAdd(&denom[r * NN + d], a);
}

// ---------------- WMMA GEMM: hs = x @ W (N x 128 @ 128 x 128, fp32) --------
// 32 rows of x per workgroup; W staged in 32x128 K-panels, double-buffered via
// global_load_async_to_lds_b128 so the DMA of panel kp+1 overlaps WMMA on kp.
constexpr int GEMM_ROWS  = 32;
constexpr int LDS_STRIDE = 132;  // padded row stride (conflict-free 16-lane rows)
constexpr int PANEL      = 32 * LDS_STRIDE;

__global__ __launch_bounds__(256)
void gemm_xw_kernel(const float* __restrict__ x, const float* __restrict__ W,
                    float* __restrict__ out) {
  __shared__ float xs[GEMM_ROWS * LDS_STRIDE];  // 32 x 128 x-tile   (16.5 KB)
  __shared__ float wsm[2 * PANEL];              // 2 x 32 x 128 W    (33.0 KB)

  const int tid  = threadIdx.x;
  const int wave = tid >> 5;          // 0..7 -> 16-wide N tile
  const int lane = tid & 31;
  const int lm   = lane & 15;         // lane % 16
  const int lh   = lane >> 4;         // lane / 16
  const int row0 = blockIdx.x * GEMM_ROWS;
  const int n0   = wave * 16;

  // ---- prologue: async-stage x-tile and W panel 0 (16B chunks per lane) ----
#pragma unroll
  for (int i = 0; i < 4; ++i) {
    const int idx = tid + i * 256;          // b128 slot in 32 x 32 grid
    const int rr  = idx >> 5;
    const int c4  = idx & 31;
    int gr = row0 + rr; if (gr >= NN) gr = NN - 1;   // clamp; stores guarded
    async_copy_b128(&xs[rr * LDS_STRIDE + c4 * 4], x + (size_t)gr * DD + c4 * 4);
  }
#pragma unroll
  for (int i = 0; i < 4; ++i) {
    const int idx = tid + i * 256;
    const int rr  = idx >> 5;
    const int c4  = idx & 31;
    async_copy_b128(&wsm[rr * LDS_STRIDE + c4 * 4], W + (size_t)rr * DD + c4 * 4);
  }
  wait_async0();
  __syncthreads();

  v8f acc[2] = {v8f{0.f}, v8f{0.f}};

  for (int kp = 0; kp < 4; ++kp) {          // 4 K-panels of 32
    // issue DMA for next panel into the other buffer (overlaps with WMMA)
    if (kp < 3) {
      float* nb = &wsm[((kp + 1) & 1) * PANEL];
#pragma unroll
      for (int i = 0; i < 4; ++i) {
        const int idx = tid + i * 256;
        const int rr  = idx >> 5;
        const int c4  = idx & 31;
        async_copy_b128(&nb[rr * LDS_STRIDE + c4 * 4],
                        W + (size_t)((kp + 1) * 32 + rr) * DD + c4 * 4);
      }
    }

    const float* wb = &wsm[(kp & 1) * PANEL];
    const int kbase = kp * 32;
#pragma unroll
    for (int kk = 0; kk < 32; kk += 4) {
      // B fragment 4x16 (ISA layout: K = vgpr + 2*(lane/16), N = lane%16)
      v2f b;
      b.x = wb[(kk + 2 * lh)     * LDS_STRIDE + n0 + lm];
      b.y = wb[(kk + 2 * lh + 1) * LDS_STRIDE + n0 + lm];
#pragma unroll
      for (int s = 0; s < 2; ++s) {
        // A fragment 16x4 (ISA layout: M = lane%16, K = vgpr + 2*(lane/16))
        const float* arow =
            &xs[(s * 16 + lm) * LDS_STRIDE + kbase + kk + 2 * lh];
        v2f a; a.x = arow[0]; a.y = arow[1];
        acc[s] = __builtin_amdgcn_wmma_f32_16x16x4_f32(
            false, a, false, b, (short)0, acc[s], false, false);
      }
    }

    wait_async0();       // next panel fully in LDS
    __syncthreads();     // nobody still reading the buffer we overwrite next
  }

  // C/D layout: row M = vgpr + 8*(lane/16), col N = lane%16
#pragma unroll
  for (int s = 0; s < 2; ++s) {
#pragma unroll
    for (int g = 0; g < 8; ++g) {
      const int gr = row0 + s * 16 + g + 8 * lh;
      if (gr < NN) out[(size_t)gr * DD + n0 + lm] = acc[s][g];
    }
  }
}

// ---------------- edge pass 3: acc[dst] += (a/denom[dst]) * hs[src] --------
__global__ __launch_bounds__(256)
void scatter_kernel(const int* __restrict__ esrc_r, const int* __restrict__ edst_r,
                    const float* __restrict__ aEdge_r, const float* __restrict__ denom_r,
                    const float* __restrict__ hs, float* __restrict__ acc) {
  const int wave = threadIdx.x >> 5;
  const int lane = threadIdx.x & 31;
  const int e = blockIdx.x * 8 + wave;
  if (e >= EE) return;
  const int s = esrc_r[e];
  const int d = edst_r[e];
  const float w = aEdge_r[e] / denom_r[d];
  const float4 hv = reinterpret_cast<const float4*>(hs + (size_t)s * DD)[lane];
  float* o = acc + (size_t)d * DD + lane * 4;
  unsafeAtomicAdd(o + 0, w * hv.x);
  unsafeAtomicAdd(o + 1, w * hv.y);
  unsafeAtomicAdd(o + 2, w * hv.z);
  unsafeAtomicAdd(o + 3, w * hv.w);
}

// ---------------- fused (relu) + LayerNorm, wave32 per node ----------------
__global__ __launch_bounds__(256)
void finalize_kernel(const float* __restrict__ acc, const float* __restrict__ gamma,
                     const float* __restrict__ beta, float* __restrict__ out,
                     int do_relu) {
  const int wave = threadIdx.x >> 5;
  const int lane = threadIdx.x & 31;
  const int node = blockIdx.x * 8 + wave;
  if (node >= NN) return;
  float4 v = reinterpret_cast<const float4*>(acc + (size_t)node * DD)[lane];
  if (do_relu) {
    v.x = fmaxf(v.x, 0.f); v.y = fmaxf(v.y, 0.f);
    v.z = fmaxf(v.z, 0.f); v.w = fmaxf(v.w, 0.f);
  }
  float s = v.x + v.y + v.z + v.w;
#pragma unroll
  for (int off = 16; off; off >>= 1) s += __shfl_xor(s, off, 32);
  const float mu = s * (1.0f / DD);
  const float dx = v.x - mu, dy = v.y - mu, dz = v.z - mu, dw = v.w - mu;
  float q = dx * dx + dy * dy + dz * dz + dw * dw;
#pragma unroll
  for (int off = 16; off; off >>= 1) q += __shfl_xor(q, off, 32);
  const float rs = rsqrtf(q * (1.0f / DD) + LN_EPS);
  const float4 g = reinterpret_cast<const float4*>(gamma)[lane];
  const float4 b = reinterpret_cast<const float4*>(beta)[lane];
  float4 o;
  o.x = dx * rs * g.x + b.x;
  o.y = dy * rs * g.y + b.y;
  o.z = dz * rs * g.z + b.z;
  o.w = dw * rs * g.w + b.w;
  reinterpret_cast<float4*>(out + (size_t)node * DD)[lane] = o;
}

// ---------------------------------------------------------------------------
extern "C" void kernel_launch(void* const* d_in, const int* in_sizes, int n_in,
                              void* d_out, int out_size, void* d_ws, size_t ws_size,
                              hipStream_t stream) {
  (void)in_sizes; (void)n_in; (void)out_size; (void)ws_size;

  const float* h        = (const float*)d_in[0];
  const int*   edge_src = (const int*)d_in[1];   // [R, E]
  const int*   edge_dst = (const int*)d_in[2];   // [R, E]
  const float* W_src    = (const float*)d_in[3]; // [L, R, D, D]
  const float* W_dst    = (const float*)d_in[4]; // [L, R, D, D]
  const float* attn_l   = (const float*)d_in[5]; // [L, R, D]
  const float* attn_r   = (const float*)d_in[6]; // [L, R, D]
  const float* gat_bias = (const float*)d_in[7]; // [L, R, D]
  const float* ntype_b  = (const float*)d_in[8]; // [L, D]
  const float* ln_g     = (const float*)d_in[9]; // [L, D]
  const float* ln_b     = (const float*)d_in[10];// [L, D]

  // -------- carve workspace (~110 MB) --------
  char* ws = (char*)d_ws;
  size_t off = 0;
  auto carve = [&](size_t nbytes) -> void* {
    void* p = ws + off;
    off += (nbytes + 255) & ~(size_t)255;
    return p;
  };
  float*    xA    = (float*)carve(sizeof(float) * (size_t)NN * DD);
  float*    xB    = (float*)carve(sizeof(float) * (size_t)NN * DD);
  float*    hs    = (float*)carve(sizeof(float) * (size_t)NN * DD);
  float*    accb  = (float*)carve(sizeof(float) * (size_t)NN * DD);
  float*    eEdge = (float*)carve(sizeof(float) * (size_t)RR * EE);
  float*    el    = (float*)carve(sizeof(float) * (size_t)RR * NN);
  float*    er    = (float*)carve(sizeof(float) * (size_t)RR * NN);
  unsigned* mEnc  = (unsigned*)carve(sizeof(unsigned) * (size_t)RR * NN);
  float*    denom = (float*)carve(sizeof(float) * (size_t)RR * NN);
  float*    wvec  = (float*)carve(sizeof(float) * (size_t)RR * DD);
  float*    uvec  = (float*)carve(sizeof(float) * (size_t)RR * DD);

  const float* x = h;
  for (int l = 0; l < LL; ++l) {
    const float* Wsl = W_src + (size_t)l * RR * DD * DD;
    const float* Wdl = W_dst + (size_t)l * RR * DD * DD;

    attn_vec_kernel<<<RR, DD, 0, stream>>>(Wsl, Wdl, attn_l + (size_t)l * RR * DD,
                                           attn_r + (size_t)l * RR * DD, wvec, uvec);
    eler_kernel<<<(NN + 7) / 8, 256, 0, stream>>>(x, wvec, uvec, el, er);
    init_acc_kernel<<<(NN * DD + 255) / 256, 256, 0, stream>>>(
        ntype_b + (size_t)l * DD, gat_bias + (size_t)l * RR * DD, accb);
    init_stats_kernel<<<(RR * NN + 255) / 256, 256, 0, stream>>>(mEnc, denom);
    edge_max_kernel<<<(RR * EE + 255) / 256, 256, 0, stream>>>(edge_src, edge_dst,
                                                               el, er, eEdge, mEnc);
    edge_exp_kernel<<<(RR * EE + 255) / 256, 256, 0, stream>>>(edge_dst, mEnc,
                                                               eEdge, denom);
    for (int r = 0; r < RR; ++r) {
      gemm_xw_kernel<<<(NN + GEMM_ROWS - 1) / GEMM_ROWS, 256, 0, stream>>>(
          x, Wsl + (size_t)r * DD * DD, hs);
      scatter_kernel<<<(EE + 7) / 8, 256, 0, stream>>>(
          edge_src + (size_t)r * EE, edge_dst + (size_t)r * EE,
          eEdge + (size_t)r * EE, denom + (size_t)r * NN, hs, accb);
    }
    float* dst = (l == LL - 1) ? (float*)d_out : ((l == 0) ? xA : xB);
    finalize_kernel<<<(NN + 7) / 8, 256, 0, stream>>>(
        accb, ln_g + (size_t)l * DD, ln_b + (size_t)l * DD, dst,
        (l < LL - 1) ? 1 : 0);
    x = dst;
  }
}